// ProposalSelectorLayer_53815940219178
// MI455X (gfx1250) — compile-verified
//
#include <hip/hip_runtime.h>

// ProposalSelectorLayer for MI455X (gfx1250, wave32).
// Pipeline: decode(+argmax partials) -> 100x [pick(1 block) -> suppress(+next partials)].
// Working set (6.3 MB SoA) lives in the 192MB L2 across all NMS iterations.

#define NPROP 262144
#define NCLS  81
#define BLK   256
#define NBLK  (NPROP / BLK)   // 1024 blocks; also the number of argmax partials
#define NMS_T 100

#define WS_Y1   0
#define WS_X1   ((size_t)NPROP)
#define WS_Y2   (2*(size_t)NPROP)
#define WS_X2   (3*(size_t)NPROP)
#define WS_AREA (4*(size_t)NPROP)
#define WS_SC   (5*(size_t)NPROP)
#define WS_PART (6*(size_t)NPROP)            // NBLK float2 pairs (val, idx-bits)
#define WS_PICK (6*(size_t)NPROP + 2*NBLK)   // [y1,x1,y2,x2,area,valid]

#ifndef USE_ASYNC_LDS
#if defined(__has_builtin)
#if __has_builtin(__builtin_amdgcn_global_load_async_to_lds_b64) && \
    __has_builtin(__builtin_amdgcn_s_wait_asynccnt)
#define USE_ASYNC_LDS 1
#endif
#endif
#endif
#ifndef USE_ASYNC_LDS
#define USE_ASYNC_LDS 0
#endif

#if USE_ASYNC_LDS
// Param 0 of the builtin is 'int2-vector __device__(AS1) *' per clang's
// diagnostic; mirror that exactly and use AS3 for the LDS destination.
typedef int v2i __attribute__((ext_vector_type(2)));
typedef __attribute__((address_space(1))) v2i g_v2i;
typedef __attribute__((address_space(3))) v2i l_v2i;
#endif

__device__ __forceinline__ float neg_inf() { return -__builtin_inff(); }

// Block-wide argmax (value, global index), tie-break to the LOWEST index
// (matches jnp.argmax first-occurrence semantics). Result -> part[blockIdx].
__device__ __forceinline__ void block_argmax_store(float v, int idx,
                                                   float2* __restrict__ part, int b) {
  __shared__ float sv[BLK];
  __shared__ int   si[BLK];
  const int tid = threadIdx.x;
  sv[tid] = v; si[tid] = idx;
  __syncthreads();
  for (int off = BLK / 2; off > 0; off >>= 1) {
    if (tid < off) {
      float v2 = sv[tid + off]; int i2 = si[tid + off];
      if (v2 > sv[tid] || (v2 == sv[tid] && i2 < si[tid])) { sv[tid] = v2; si[tid] = i2; }
    }
    __syncthreads();
  }
  if (tid == 0) part[b] = make_float2(sv[0], __int_as_float(si[0]));
}

__global__ void decode_kernel(const float* __restrict__ prop,
                              const float* __restrict__ delt,
                              const float* __restrict__ cls,
                              const int* __restrict__ img_h,
                              const int* __restrict__ img_w,
                              float* __restrict__ ws) {
  const int i = blockIdx.x * BLK + threadIdx.x;           // exact cover of NPROP
  const float H = (float)*img_h;
  const float W = (float)*img_w;

  // --- class max (argmax value only is needed) -----------------------------
  const float* row = cls + (size_t)i * NCLS;
  // gfx1250 speculative prefetch of the next tile's rows (global_prefetch_b8)
  if (i + BLK < NPROP) __builtin_prefetch(cls + (size_t)(i + BLK) * NCLS, 0, 1);
  float m = row[0];
#pragma unroll 3
  for (int j = 1; j < NCLS; ++j) m = fmaxf(m, row[j]);

  // --- box decode ----------------------------------------------------------
  const float4 p = reinterpret_cast<const float4*>(prop)[i];
  const float4 d = reinterpret_cast<const float4*>(delt)[i];
  const float y1p = p.x * H, x1p = p.y * W, h1 = p.z * H, w1 = p.w * W;
  const float d0 = tanhf(d.x), d1 = tanhf(d.y), d2 = tanhf(d.z), d3 = tanhf(d.w);
  const float yy = d0 * h1 + y1p;
  const float xx = d1 * w1 + x1p;
  const float h2 = (d2 + 1.0f) * h1;
  const float w2 = (d3 + 1.0f) * w1;
  const float b0 = fminf(fmaxf(yy,       0.0f), H);
  const float b1 = fminf(fmaxf(xx,       0.0f), W);
  const float b2 = fminf(fmaxf(yy + h2,  0.0f), H);
  const float b3 = fminf(fmaxf(xx + w2,  0.0f), W);
  const float area = (b2 - b0) * (b3 - b1);
  const float sc = (m > 0.7f) ? m : neg_inf();

  ws[WS_Y1 + i] = b0;
  ws[WS_X1 + i] = b1;
  ws[WS_Y2 + i] = b2;
  ws[WS_X2 + i] = b3;
  ws[WS_AREA + i] = area;
  ws[WS_SC + i] = sc;

  // partials for NMS iteration 0
  block_argmax_store(sc, i, (float2*)(ws + WS_PART), blockIdx.x);
}

// Single block: reduce NBLK partials -> global argmax, emit output row t,
// publish pick descriptor, kill the picked score.
__global__ void pick_kernel(float* __restrict__ ws, float* __restrict__ out, int t) {
  __shared__ float2 sp[NBLK];                 // 8 KB staged partials
  const int tid = threadIdx.x;
  const float2* gp = (const float2*)(ws + WS_PART);

#if USE_ASYNC_LDS
  // CDNA5 async global->LDS copy path (ASYNCcnt-tracked, no VGPR round trip)
  for (int k = tid; k < NBLK; k += BLK) {
    __builtin_amdgcn_global_load_async_to_lds_b64(
        (g_v2i*)(void*)(gp + k), (l_v2i*)(sp + k), 0, 0);
  }
  __builtin_amdgcn_s_wait_asynccnt(0);
  __syncthreads();
#else
  for (int k = tid; k < NBLK; k += BLK) sp[k] = gp[k];
  __syncthreads();
#endif

  float bv = neg_inf();
  int   bi = 0x7fffffff;
  for (int k = tid; k < NBLK; k += BLK) {
    const float v = sp[k].x;
    const int  ix = __float_as_int(sp[k].y);
    if (v > bv || (v == bv && ix < bi)) { bv = v; bi = ix; }
  }

  __shared__ float sv[BLK];
  __shared__ int   si[BLK];
  sv[tid] = bv; si[tid] = bi;
  __syncthreads();
  for (int off = BLK / 2; off > 0; off >>= 1) {
    if (tid < off) {
      float v2 = sv[tid + off]; int i2 = si[tid + off];
      if (v2 > sv[tid] || (v2 == sv[tid] && i2 < si[tid])) { sv[tid] = v2; si[tid] = i2; }
    }
    __syncthreads();
  }

  if (tid == 0) {
    const int   i = si[0];
    const float v = sv[0];
    const int valid = (v != neg_inf()) ? 1 : 0;   // scores are finite or -inf
    const float y1 = ws[WS_Y1 + i], x1 = ws[WS_X1 + i];
    const float y2 = ws[WS_Y2 + i], x2 = ws[WS_X2 + i];
    const float ar = ws[WS_AREA + i];

    float* pk = ws + WS_PICK;
    pk[0] = y1; pk[1] = x1; pk[2] = y2; pk[3] = x2; pk[4] = ar;
    pk[5] = valid ? 1.0f : 0.0f;

    // outputs: boxes[100][4] | scores[100] | valid[100]
    out[4 * t + 0] = valid ? y1 : 0.0f;
    out[4 * t + 1] = valid ? x1 : 0.0f;
    out[4 * t + 2] = valid ? y2 : 0.0f;
    out[4 * t + 3] = valid ? x2 : 0.0f;
    out[4 * NMS_T + t] = valid ? v : 0.0f;
    out[5 * NMS_T + t] = valid ? 1.0f : 0.0f;

    ws[WS_SC + i] = neg_inf();                 // scores.at[i].set(-inf)
  }
}

// Suppress IoU>0.5 vs the current pick, then produce the NEXT iteration's
// per-block argmax partials (fused to halve launch count).
__global__ void suppress_kernel(float* __restrict__ ws) {
  const int i = blockIdx.x * BLK + threadIdx.x;
  const float* pk = ws + WS_PICK;
  const float py1 = pk[0], px1 = pk[1], py2 = pk[2], px2 = pk[3], pa = pk[4];
  const bool pvalid = (pk[5] != 0.0f);

  float s = ws[WS_SC + i];
  if (pvalid) {
    const float y1 = ws[WS_Y1 + i], x1 = ws[WS_X1 + i];
    const float y2 = ws[WS_Y2 + i], x2 = ws[WS_X2 + i];
    const float ar = ws[WS_AREA + i];
    const float iy1 = fmaxf(py1, y1), ix1 = fmaxf(px1, x1);
    const float iy2 = fminf(py2, y2), ix2 = fminf(px2, x2);
    const float inter = fmaxf(iy2 - iy1, 0.0f) * fmaxf(ix2 - ix1, 0.0f);
    const float iou = inter / (pa + ar - inter + 1e-9f);
    if (iou > 0.5f) { s = neg_inf(); ws[WS_SC + i] = s; }
  }
  block_argmax_store(s, i, (float2*)(ws + WS_PART), blockIdx.x);
}

extern "C" void kernel_launch(void* const* d_in, const int* in_sizes, int n_in,
                              void* d_out, int out_size, void* d_ws, size_t ws_size,
                              hipStream_t stream) {
  const float* prop = (const float*)d_in[0];
  const float* delt = (const float*)d_in[1];
  const float* cls  = (const float*)d_in[2];
  const int*   ih   = (const int*)d_in[3];
  const int*   iw   = (const int*)d_in[4];
  float* out = (float*)d_out;
  float* ws  = (float*)d_ws;   // needs (6*N + 2*NBLK + 8) floats ~= 6.3 MB

  decode_kernel<<<NBLK, BLK, 0, stream>>>(prop, delt, cls, ih, iw, ws);
  for (int t = 0; t < NMS_T; ++t) {
    pick_kernel<<<1, BLK, 0, stream>>>(ws, out, t);
    suppress_kernel<<<NBLK, BLK, 0, stream>>>(ws);
  }
}